// GraphSAGE_65721589563777
// MI455X (gfx1250) — compile-verified
//
#include <hip/hip_runtime.h>

// CDNA5 / gfx1250, wave32. GraphSAGE 3-layer forward.
// f32 end-to-end; GEMMs use V_WMMA_F32_16X16X4_F32 (exact f32 MACs).

typedef float v2f __attribute__((ext_vector_type(2)));
typedef float v8f __attribute__((ext_vector_type(8)));

#define HID 128

// ---------------------------------------------------------------- degree
__global__ void deg_kernel(const int* __restrict__ dst, float* __restrict__ deg, int E) {
    int e = blockIdx.x * blockDim.x + threadIdx.x;
    if (e < E) unsafeAtomicAdd(&deg[dst[e]], 1.0f);
}

__global__ void invdeg_kernel(float* __restrict__ deg, int N) {
    int i = blockIdx.x * blockDim.x + threadIdx.x;
    if (i < N) deg[i] = 1.0f / fmaxf(deg[i], 1.0f);
}

// ------------------------------------------------- edge scatter (sum agg)
// one thread = one (edge, 4-feature chunk); 32 chunks cover 128 features
__global__ void scatter_kernel(const float* __restrict__ X,
                               const int* __restrict__ src,
                               const int* __restrict__ dst,
                               float* __restrict__ agg, int E) {
    int idx = blockIdx.x * blockDim.x + threadIdx.x;
    int e = idx >> 5;
    if (e >= E) return;
    int c = (idx & 31) << 2;
    int s = src[e], t = dst[e];
    float4 v = *(const float4*)(X + (size_t)s * HID + c);
    float* out = agg + (size_t)t * HID + c;
    unsafeAtomicAdd(out + 0, v.x);
    unsafeAtomicAdd(out + 1, v.y);
    unsafeAtomicAdd(out + 2, v.z);
    unsafeAtomicAdd(out + 3, v.w);
}

// ------------------------------------------------------ fused SAGE GEMM
// Y[16 rows x DOUT] = relu?( (AGG*invdeg) @ Wl + b + X @ Wr )
// block = (DOUT/16) waves; wave w -> output cols [16w, 16w+16)
// V_WMMA_F32_16X16X4_F32 operand mapping (wave32):
//   A 16x4 : VGPR j, lane-half h -> K = j + 2h ; M = lane&15
//   B 4x16 : VGPR j, lane-half h -> K = j + 2h ; N = lane&15
//   C 16x16: VGPR v -> M = v + 8*(lane>>4)     ; N = lane&15
template <int DOUT, bool RELU>
__global__ __launch_bounds__((DOUT / 16) * 32)
void sage_gemm(const float* __restrict__ X, const float* __restrict__ AGG,
               const float* __restrict__ invdeg,
               const float* __restrict__ Wl, const float* __restrict__ Wr,
               const float* __restrict__ bias, float* __restrict__ Y, int N) {
    __shared__ float Xs[16][HID + 4];   // +4 pad -> conflict-free column reads
    __shared__ float As[16][HID + 4];

    const int tid = threadIdx.x;
    const int nth = (DOUT / 16) * 32;
    const int row_base = blockIdx.x * 16;

    // stage the 16x128 X tile and deg-scaled AGG tile into LDS (float4)
    for (int i = tid; i < 16 * (HID / 4); i += nth) {
        int row = i >> 5;
        int c = (i & 31) << 2;
        int gr = row_base + row;
        if (gr > N - 1) gr = N - 1;              // clamp (N divisible by 16 here)
        size_t g = (size_t)gr * HID + c;
        *(float4*)&Xs[row][c] = *(const float4*)(X + g);
        float sc = invdeg[gr];
        float4 a = *(const float4*)(AGG + g);
        a.x *= sc; a.y *= sc; a.z *= sc; a.w *= sc;
        *(float4*)&As[row][c] = a;
    }
    __syncthreads();

    const int lane = tid & 31;
    const int wave = tid >> 5;
    const int half = lane >> 4;
    const int r = lane & 15;
    const int nb = wave * 16 + r;               // output column of this lane

    v8f c;
    const float bv = bias[nb];
#pragma unroll
    for (int v = 0; v < 8; ++v) c[v] = bv;      // bias folded into accumulator

#pragma unroll
    for (int k0 = 0; k0 < HID; k0 += 4) {
        const int kk = k0 + half * 2;
        // aggr @ Wl
        v2f a0 = {As[r][kk], As[r][kk + 1]};
        v2f b0 = {Wl[kk * DOUT + nb], Wl[(kk + 1) * DOUT + nb]};
        c = __builtin_amdgcn_wmma_f32_16x16x4_f32(false, a0, false, b0,
                                                  (short)0, c, false, false);
        // x_root @ Wr
        v2f a1 = {Xs[r][kk], Xs[r][kk + 1]};
        v2f b1 = {Wr[kk * DOUT + nb], Wr[(kk + 1) * DOUT + nb]};
        c = __builtin_amdgcn_wmma_f32_16x16x4_f32(false, a1, false, b1,
                                                  (short)0, c, false, false);
    }

#pragma unroll
    for (int v = 0; v < 8; ++v) {
        int row = row_base + v + 8 * half;
        if (row < N) {
            float val = c[v];
            if (RELU) val = fmaxf(val, 0.0f);
            Y[(size_t)row * DOUT + nb] = val;
        }
    }
}

// ----------------------------------------------------------------- driver
extern "C" void kernel_launch(void* const* d_in, const int* in_sizes, int n_in,
                              void* d_out, int out_size, void* d_ws, size_t ws_size,
                              hipStream_t stream) {
    const float* x   = (const float*)d_in[0];
    const int*   ei  = (const int*)d_in[1];     // [2, E] int32: row0=src, row1=dst
    const float* wl0 = (const float*)d_in[2];
    const float* b0  = (const float*)d_in[3];
    const float* wr0 = (const float*)d_in[4];
    const float* wl1 = (const float*)d_in[5];
    const float* b1  = (const float*)d_in[6];
    const float* wr1 = (const float*)d_in[7];
    const float* wl2 = (const float*)d_in[8];
    const float* b2  = (const float*)d_in[9];
    const float* wr2 = (const float*)d_in[10];

    const int N = in_sizes[0] / HID;
    const int E = in_sizes[1] / 2;
    const int* src = ei;
    const int* dst = ei + E;

    // workspace carve-up: invdeg | agg | h0 | h1
    float* ws = (float*)d_ws;
    size_t NP = ((size_t)N + 63) & ~(size_t)63;
    float* invdeg = ws;
    float* agg    = ws + NP;
    float* h0     = agg + (size_t)N * HID;
    float* h1     = h0 + (size_t)N * HID;

    const int row_blocks = (N + 15) / 16;
    const long scat_work = (long)E * 32;
    const int scat_blocks = (int)((scat_work + 255) / 256);
    const size_t agg_bytes = (size_t)N * HID * sizeof(float);

    // degrees (shared by all layers)
    hipMemsetAsync(invdeg, 0, (size_t)N * sizeof(float), stream);
    deg_kernel<<<(E + 255) / 256, 256, 0, stream>>>(dst, invdeg, E);
    invdeg_kernel<<<(N + 255) / 256, 256, 0, stream>>>(invdeg, N);

    // layer 0: x -> h0 (relu)
    hipMemsetAsync(agg, 0, agg_bytes, stream);
    scatter_kernel<<<scat_blocks, 256, 0, stream>>>(x, src, dst, agg, E);
    sage_gemm<128, true><<<row_blocks, 256, 0, stream>>>(x, agg, invdeg, wl0, wr0, b0, h0, N);

    // layer 1: h0 -> h1 (relu)
    hipMemsetAsync(agg, 0, agg_bytes, stream);
    scatter_kernel<<<scat_blocks, 256, 0, stream>>>(h0, src, dst, agg, E);
    sage_gemm<128, true><<<row_blocks, 256, 0, stream>>>(h0, agg, invdeg, wl1, wr1, b1, h1, N);

    // layer 2: h1 -> out (no relu, DOUT=64)
    hipMemsetAsync(agg, 0, agg_bytes, stream);
    scatter_kernel<<<scat_blocks, 256, 0, stream>>>(h1, src, dst, agg, E);
    sage_gemm<64, false><<<row_blocks, 128, 0, stream>>>(h1, agg, invdeg, wl2, wr2, b2,
                                                         (float*)d_out, N);
}